// AlphaCompositor_9268539424960
// MI455X (gfx1250) — compile-verified
//
#include <hip/hip_runtime.h>
#include <stdint.h>

// Problem constants (match reference.py)
#define NN 8
#define KK 8
#define HH 512
#define WW 512
#define CC 4
#define PP 100000
#define HWX (HH * WW)          // 262144 pixels per (n,k) plane

#define TPB 256                // 8 wave32 per block
#define PPT 4                  // pixels per thread (B128 granularity)
#define TILE (TPB * PPT)       // 1024 pixels per block; HWX % TILE == 0

// ---------------------------------------------------------------------------
// CDNA5 async global->LDS copy (ASYNCcnt path). Staging through LDS costs no
// data VGPRs for the 16 in-flight B128 loads, so occupancy stays high.
// ---------------------------------------------------------------------------
__device__ __forceinline__ uint32_t lds_addr32(const void* p) {
  // low 32 bits of the flat address of an LDS object == wave-relative LDS
  // byte address (aperture bits live in [63:32])
  return (uint32_t)(uintptr_t)p;
}

__device__ __forceinline__ void async_load_b128(uint32_t lds_off, const void* gaddr) {
  asm volatile("global_load_async_to_lds_b128 %0, %1, off"
               :
               : "v"(lds_off), "v"(gaddr)
               : "memory");
}

__device__ __forceinline__ void wait_async0() {
#if __has_builtin(__builtin_amdgcn_s_wait_asynccnt)
  __builtin_amdgcn_s_wait_asynccnt(0);
#else
  asm volatile("s_wait_asynccnt 0" ::: "memory");
#endif
}

// ---------------------------------------------------------------------------
// Pre-pass: transpose ptclds [C=4, P] -> tex [P] of float4 so each point
// gather is a single 16B L2 transaction instead of 4 scattered 4B ones.
// ---------------------------------------------------------------------------
__global__ __launch_bounds__(256) void transpose_feats(const float* __restrict__ ptclds,
                                                       float4* __restrict__ tex) {
  int p = blockIdx.x * 256 + threadIdx.x;
  if (p < PP) {
    tex[p] = make_float4(ptclds[0 * PP + p], ptclds[1 * PP + p],
                         ptclds[2 * PP + p], ptclds[3 * PP + p]);
  }
}

// ---------------------------------------------------------------------------
// Main compositor. One block = 1024 contiguous pixels of one image n
// (HWX % TILE == 0, so tiles never straddle images).
// out layout: images [N,C,H,W] (float) followed by mask [N,H,W] (float 1/0).
// ---------------------------------------------------------------------------
template <bool USE_TEX>
__global__ __launch_bounds__(TPB) void composite_kernel(
    const int* __restrict__ pix_idxs,    // [N,K,H,W]
    const float* __restrict__ alphas,    // [N,K,H,W]
    const float* __restrict__ ptclds,    // [C,P]
    const float4* __restrict__ tex,      // [P], valid iff USE_TEX
    float* __restrict__ out) {
  __shared__ int4   sm_idx[KK][TPB];     // 32 KB
  __shared__ float4 sm_alp[KK][TPB];     // 32 KB

  const int tid = threadIdx.x;
  const long long pix_base = (long long)blockIdx.x * TILE;
  const int n   = (int)(pix_base / HWX);
  const int hw0 = (int)(pix_base % HWX) + tid * PPT;   // 16B-aligned element offset

  // Stage all K planes for this thread's 4 pixels: 16 async B128 copies.
#pragma unroll
  for (int k = 0; k < KK; ++k) {
    const size_t goff = ((size_t)(n * KK + k)) * HWX + (size_t)hw0;
    async_load_b128(lds_addr32(&sm_idx[k][tid]), pix_idxs + goff);
    async_load_b128(lds_addr32(&sm_alp[k][tid]), alphas + goff);
  }
  wait_async0();   // s_wait_asynccnt 0 -> data resident in LDS

  float4 acc[PPT];
  float  T[PPT];
  bool   fg[PPT];
#pragma unroll
  for (int j = 0; j < PPT; ++j) {
    acc[j] = make_float4(0.f, 0.f, 0.f, 0.f);
    T[j] = 1.f;
    fg[j] = true;
  }

#pragma unroll
  for (int k = 0; k < KK; ++k) {
    const int4   id4 = sm_idx[k][tid];
    const float4 a4  = sm_alp[k][tid];
    const int   ids[PPT] = {id4.x, id4.y, id4.z, id4.w};
    const float as_[PPT] = {a4.x, a4.y, a4.z, a4.w};
#pragma unroll
    for (int j = 0; j < PPT; ++j) {
      const int  id    = ids[j];
      const bool valid = (id >= 0);
      if (k == 0) fg[j] = valid;
      const float a = valid ? as_[j] : 0.f;   // trailing -1 padding => alpha 0
      const float w = a * T[j];               // a_k * prod_{j<k}(1-a_j)
      T[j] = T[j] * (1.f - a);
      const int sid = valid ? id : 0;         // clamped gather, w==0 kills it
      float fx, fy, fz, fw;
      if (USE_TEX) {
        const float4 f = tex[sid];            // single B128 L2 gather
        fx = f.x; fy = f.y; fz = f.z; fw = f.w;
      } else {
        fx = ptclds[0 * PP + sid];
        fy = ptclds[1 * PP + sid];
        fz = ptclds[2 * PP + sid];
        fw = ptclds[3 * PP + sid];
      }
      acc[j].x += w * fx;
      acc[j].y += w * fy;
      acc[j].z += w * fz;
      acc[j].w += w * fw;
    }
  }

  // Background override: images[bg] = (0,0,0,1); mask = foreground.
#pragma unroll
  for (int j = 0; j < PPT; ++j) {
    if (!fg[j]) acc[j] = make_float4(0.f, 0.f, 0.f, 1.f);
  }

  // Store images [N,C,H,W]: transpose pixel-major acc -> channel-major float4.
  const size_t cbase = ((size_t)n * CC) * HWX + (size_t)hw0;
  const float4 v0 = make_float4(acc[0].x, acc[1].x, acc[2].x, acc[3].x);
  const float4 v1 = make_float4(acc[0].y, acc[1].y, acc[2].y, acc[3].y);
  const float4 v2 = make_float4(acc[0].z, acc[1].z, acc[2].z, acc[3].z);
  const float4 v3 = make_float4(acc[0].w, acc[1].w, acc[2].w, acc[3].w);
  *reinterpret_cast<float4*>(out + cbase + 0 * (size_t)HWX) = v0;
  *reinterpret_cast<float4*>(out + cbase + 1 * (size_t)HWX) = v1;
  *reinterpret_cast<float4*>(out + cbase + 2 * (size_t)HWX) = v2;
  *reinterpret_cast<float4*>(out + cbase + 3 * (size_t)HWX) = v3;

  // Mask (float 1/0) appended after the images block.
  float* mask = out + (size_t)NN * CC * HWX;
  const float4 m = make_float4(fg[0] ? 1.f : 0.f, fg[1] ? 1.f : 0.f,
                               fg[2] ? 1.f : 0.f, fg[3] ? 1.f : 0.f);
  *reinterpret_cast<float4*>(mask + (size_t)n * HWX + (size_t)hw0) = m;
}

// ---------------------------------------------------------------------------
extern "C" void kernel_launch(void* const* d_in, const int* in_sizes, int n_in,
                              void* d_out, int out_size, void* d_ws, size_t ws_size,
                              hipStream_t stream) {
  (void)in_sizes; (void)n_in; (void)out_size;
  const int*   pix = (const int*)d_in[0];
  const float* alp = (const float*)d_in[1];
  const float* pts = (const float*)d_in[2];
  float* out = (float*)d_out;

  const int nblocks = (NN * HWX) / TILE;  // 2048

  if (ws_size >= (size_t)PP * sizeof(float4)) {
    float4* tex = (float4*)d_ws;
    transpose_feats<<<(PP + 255) / 256, 256, 0, stream>>>(pts, tex);
    composite_kernel<true><<<nblocks, TPB, 0, stream>>>(pix, alp, pts, tex, out);
  } else {
    composite_kernel<false><<<nblocks, TPB, 0, stream>>>(pix, alp, pts, nullptr, out);
  }
}